// BiLSTMClassifier_85255100826308
// MI455X (gfx1250) — compile-verified
//
#include <hip/hip_runtime.h>

// BiLSTM classifier for MI455X (gfx1250, wave32).
//   B=512 batch, T=1000 steps, IN=4, H=64.
// Kernel 1: persistent forward LSTM. 32 workgroups x 128 threads (4 waves).
//   Each workgroup owns a 16-row batch tile for the whole sequence.
//   Gates via V_WMMA_F32_16X16X4_F32 (full f32 precision, matches reference):
//     per wave per step: 4 N-tiles x (1 x-part WMMA + 16 K-chunk h WMMAs).
//   h shared across waves through a double-buffered, bank-padded LDS tile;
//   one barrier per step. Weights pre-staged in VGPRs in B-operand layout.
// Kernel 2: backward-direction first cell (h0=c0=0 => f-gate & w_hh_b unused)
//   fused with the FC head + sigmoid.

#define HSZ  64
#define INSZ 4

typedef float v2f __attribute__((ext_vector_type(2)));
typedef float v8f __attribute__((ext_vector_type(8)));

__device__ __forceinline__ float sigmoid_f(float v) {
    return 1.0f / (1.0f + __expf(-v));
}
__device__ __forceinline__ float tanh_f(float v) {
    // robust at +/-inf: e->inf gives 1, e->0 gives -1
    float e = __expf(2.0f * v);
    return 1.0f - 2.0f / (e + 1.0f);
}

__global__ void __launch_bounds__(128, 1)
lstm_fwd_kernel(const float* __restrict__ x,
                const float* __restrict__ w_ih,   // [4H, IN]
                const float* __restrict__ w_hh,   // [4H, H]
                const float* __restrict__ b_ih,   // [4H]
                const float* __restrict__ b_hh,   // [4H]
                float* __restrict__ h_out,        // [B, H]
                int Tlen)
{
    // double-buffered h tile; +2 pad so row stride (66 dwords) breaks the
    // 64-bank alias on "all lanes read distinct rows, same column" A-reads.
    __shared__ float h_lds[2][16][HSZ + 2];

    const int tid  = threadIdx.x;
    const int lane = tid & 31;
    const int wave = tid >> 5;       // 0..3: owns cols [16w,16w+16) of each gate
    const int nl   = lane & 15;      // tile-local column (B/C) or row M (A)
    const int half = lane >> 4;      // 0/1: lane half
    const int kk   = half * 2;       // K sub-offset for 32-bit A/B layouts
    const int m0   = blockIdx.x * 16;

    // ---- stage weights in VGPRs, B-operand (4x16 KxN) layout ----
    // B element (k, n): lane = (k/2)*16 + n_local, vgpr = k%2
    v2f   whhB[4][16];   // [gate][K-chunk]
    v2f   wihB[4];       // [gate]
    float bias[4];       // b_ih + b_hh, per tile column
#pragma unroll
    for (int g = 0; g < 4; ++g) {
        const int n_abs = g * HSZ + wave * 16 + nl;    // row of [4H, *]
        const float* wr = w_hh + (size_t)n_abs * HSZ;
#pragma unroll
        for (int kc = 0; kc < 16; ++kc)
            whhB[g][kc] = *(const v2f*)(wr + kc * 4 + kk);
        wihB[g] = *(const v2f*)(w_ih + n_abs * INSZ + kk);
        bias[g] = b_ih[n_abs] + b_hh[n_abs];
    }

    // h starts at zero
    for (int i = tid; i < 16 * (HSZ + 2); i += 128)
        ((float*)h_lds[0])[i] = 0.0f;
    __syncthreads();

    v8f c_state = {};  // wave-local c tile [16,16] in C layout

    // x A-operand: lane -> row m0+nl, K elements {kk, kk+1}
    const float* xrow = x + ((size_t)(m0 + nl) * Tlen) * INSZ + kk;

    for (int t = 0; t < Tlen; ++t) {
        const int cur = t & 1;

        v2f xA = *(const v2f*)(xrow + t * INSZ);
        __builtin_prefetch(xrow + ((t + 16 < Tlen) ? (t + 16) : t) * INSZ, 0, 1);

        // gates = bias + x_t @ W_ih^T  (one 16x16x4 f32 WMMA per gate tile)
        v8f acc[4];
#pragma unroll
        for (int g = 0; g < 4; ++g) {
            v8f cinit;
#pragma unroll
            for (int r = 0; r < 8; ++r) cinit[r] = bias[g];
            acc[g] = __builtin_amdgcn_wmma_f32_16x16x4_f32(
                false, xA, false, wihB[g], (short)0, cinit, false, false);
        }

        // gates += h @ W_hh^T : 16 K-chunks of 4, h A-chunk streamed from LDS.
        // 4 independent accumulation chains (one per gate) hide WMMA latency.
#pragma unroll
        for (int kc = 0; kc < 16; ++kc) {
            v2f hA = *(const v2f*)(&h_lds[cur][nl][kc * 4 + kk]);
#pragma unroll
            for (int g = 0; g < 4; ++g)
                acc[g] = __builtin_amdgcn_wmma_f32_16x16x4_f32(
                    false, hA, false, whhB[g][kc], (short)0, acc[g], false, false);
        }

        // activations + state update, all wave-local (gate-complete tiling).
        // C layout: vgpr r -> row r + half*8, col = wave*16 + nl.
#pragma unroll
        for (int r = 0; r < 8; ++r) {
            float iv = sigmoid_f(acc[0][r]);
            float fv = sigmoid_f(acc[1][r]);
            float gv = tanh_f(acc[2][r]);
            float ov = sigmoid_f(acc[3][r]);
            float cv = fv * c_state[r] + iv * gv;
            c_state[r] = cv;
            float hv = ov * tanh_f(cv);
            h_lds[1 - cur][r + half * 8][wave * 16 + nl] = hv;
        }
        __syncthreads();   // publish h(t+1) buffer
    }

    const int fin = Tlen & 1;  // buffer holding h after the last step
    for (int i = tid; i < 16 * HSZ; i += 128)
        h_out[(size_t)(m0 + i / HSZ) * HSZ + (i % HSZ)] = h_lds[fin][i / HSZ][i % HSZ];
}

// Backward-direction LSTM is only evaluated at its first step from zero state:
//   c = sigmoid(i)*tanh(g); h = sigmoid(o)*tanh(c)  (f-gate multiplies c0=0).
// Fused with the FC head: out[b] = sigmoid(h_fwd.w_fc[:64] + h_bwd.w_fc[64:] + b_fc)
__global__ void __launch_bounds__(128)
bwd_fc_kernel(const float* __restrict__ x,
              const float* __restrict__ w_ih_b,  // [4H, IN]
              const float* __restrict__ b_ih_b,
              const float* __restrict__ b_hh_b,
              const float* __restrict__ w_fc,    // [1, 2H]
              const float* __restrict__ b_fc,    // [1]
              const float* __restrict__ h_fwd,   // [B, H]
              float* __restrict__ out,
              int B, int Tlen)
{
    int b = blockIdx.x * blockDim.x + threadIdx.x;
    if (b >= B) return;

    const float* xl = x + ((size_t)b * Tlen + (Tlen - 1)) * INSZ;
    float x0 = xl[0], x1 = xl[1], x2 = xl[2], x3 = xl[3];

    float acc = b_fc[0];
    const float* hf = h_fwd + (size_t)b * HSZ;
#pragma unroll 8
    for (int n = 0; n < HSZ; ++n)
        acc += hf[n] * w_fc[n];

    for (int n = 0; n < HSZ; ++n) {
        const float* wi = w_ih_b + (size_t)(n          ) * INSZ;
        const float* wg = w_ih_b + (size_t)(n + 2 * HSZ) * INSZ;
        const float* wo = w_ih_b + (size_t)(n + 3 * HSZ) * INSZ;
        float gi = b_ih_b[n]           + b_hh_b[n]           + x0*wi[0] + x1*wi[1] + x2*wi[2] + x3*wi[3];
        float gg = b_ih_b[n + 2*HSZ]   + b_hh_b[n + 2*HSZ]   + x0*wg[0] + x1*wg[1] + x2*wg[2] + x3*wg[3];
        float go = b_ih_b[n + 3*HSZ]   + b_hh_b[n + 3*HSZ]   + x0*wo[0] + x1*wo[1] + x2*wo[2] + x3*wo[3];
        float cv = sigmoid_f(gi) * tanh_f(gg);
        float hv = sigmoid_f(go) * tanh_f(cv);
        acc += hv * w_fc[HSZ + n];
    }
    out[b] = sigmoid_f(acc);
}

extern "C" void kernel_launch(void* const* d_in, const int* in_sizes, int n_in,
                              void* d_out, int out_size, void* d_ws, size_t ws_size,
                              hipStream_t stream)
{
    (void)n_in; (void)ws_size;
    const float* x      = (const float*)d_in[0];
    const float* w_ih_f = (const float*)d_in[1];
    const float* w_hh_f = (const float*)d_in[2];
    const float* b_ih_f = (const float*)d_in[3];
    const float* b_hh_f = (const float*)d_in[4];
    const float* w_ih_b = (const float*)d_in[5];
    // d_in[6] = w_hh_b: mathematically unused (backward step runs from h0=c0=0)
    const float* b_ih_b = (const float*)d_in[7];
    const float* b_hh_b = (const float*)d_in[8];
    const float* w_fc   = (const float*)d_in[9];
    const float* b_fc   = (const float*)d_in[10];

    float* out   = (float*)d_out;
    float* h_fwd = (float*)d_ws;                 // [B, H] f32 scratch (128 KB)

    const int B    = out_size;                   // 512
    const int Tlen = in_sizes[0] / (B * INSZ);   // 1000

    lstm_fwd_kernel<<<B / 16, 128, 0, stream>>>(
        x, w_ih_f, w_hh_f, b_ih_f, b_hh_f, h_fwd, Tlen);
    bwd_fc_kernel<<<(B + 127) / 128, 128, 0, stream>>>(
        x, w_ih_b, b_ih_b, b_hh_b, w_fc, b_fc, h_fwd, out, B, Tlen);
}